// SEConvModel_87909390614540
// MI455X (gfx1250) — compile-verified
//
#include <hip/hip_runtime.h>
#include <math.h>

typedef float v2f __attribute__((ext_vector_type(2)));
typedef float v8f __attribute__((ext_vector_type(8)));

#define H0C 64
#define H1C 32

// f32 WMMA 16x16x4: A = 16x4 (2 VGPR/lane), B = 4x16 (2 VGPR/lane), C/D = 16x16 (8 VGPR/lane)
__device__ __forceinline__ v8f wmma4(v2f a, v2f b, v8f c) {
  return __builtin_amdgcn_wmma_f32_16x16x4_f32(false, a, false, b, (short)0, c, false, false);
}

// One K-step of the scalar-input path: 6 Wss tiles (ea0-folded A) + 2 Wsv tiles.
__device__ __forceinline__ void step_scalar(int m, int kl, v2f A, v2f Ae,
                                            const float* __restrict__ WssT,
                                            const float* __restrict__ WsvT,
                                            v8f* __restrict__ accS,
                                            v8f* __restrict__ accT0) {
  #pragma unroll
  for (int nt = 0; nt < 6; ++nt) {
    const v2f B = *(const v2f*)(WssT + (size_t)(nt * 16 + m) * 132 + kl);
    accS[nt] = wmma4(Ae, B, accS[nt]);
  }
  #pragma unroll
  for (int nt = 0; nt < 2; ++nt) {
    const v2f B = *(const v2f*)(WsvT + (size_t)(nt * 16 + m) * 132 + kl);
    accT0[nt] = wmma4(A, B, accT0[nt]);
  }
}

// One K-step of the vector-input path: 6 Wvs tiles + 2x(Wvv0,Wvv1) tiles,
// with the Wvv0/Wvv1 B fragments reused across the 3 spatial components.
__device__ __forceinline__ void step_vector(int m, int j0, v2f Ad, v2f Ax, v2f Ay, v2f Az,
                                            v2f Cx, v2f Cy, v2f Cz,
                                            const float* __restrict__ WvsT,
                                            const float* __restrict__ W0T,
                                            const float* __restrict__ W1T,
                                            v8f* __restrict__ accS,
                                            v8f* __restrict__ aVx,
                                            v8f* __restrict__ aVy,
                                            v8f* __restrict__ aVz) {
  #pragma unroll
  for (int nt = 0; nt < 6; ++nt) {
    const v2f B = *(const v2f*)(WvsT + (size_t)(nt * 16 + m) * 64 + j0);
    accS[nt] = wmma4(Ad, B, accS[nt]);
  }
  #pragma unroll
  for (int nt = 0; nt < 2; ++nt) {
    const v2f B0 = *(const v2f*)(W0T + (size_t)(nt * 16 + m) * 64 + j0);
    aVx[nt] = wmma4(Ax, B0, aVx[nt]);
    aVy[nt] = wmma4(Ay, B0, aVy[nt]);
    aVz[nt] = wmma4(Az, B0, aVz[nt]);
    const v2f B1 = *(const v2f*)(W1T + (size_t)(nt * 16 + m) * 64 + j0);
    aVx[nt] = wmma4(Cx, B1, aVx[nt]);
    aVy[nt] = wmma4(Cy, B1, aVy[nt]);
    aVz[nt] = wmma4(Cz, B1, aVz[nt]);
  }
}

// ---------------------------------------------------------------------------
// Weight prep: transpose + pad into [N][Kpad] so a B fragment is one b64 load.
//   WssT [96][132] @0, WsvT [32][132] @12672, WvsT [96][64] @16896,
//   Wvv0T [32][64] @23040, Wvv1T [32][64] @25088  -> 27136 floats/layer
// ---------------------------------------------------------------------------
__global__ void prep_weights(const float* __restrict__ Wss, const float* __restrict__ Wsv,
                             const float* __restrict__ Wvs, const float* __restrict__ Wvv0,
                             const float* __restrict__ Wvv1, float* __restrict__ dst) {
  int i = blockIdx.x * blockDim.x + threadIdx.x;
  if (i < 12672) {
    int n = i / 132, k = i % 132;
    dst[i] = (k < 130) ? Wss[k * 96 + n] : 0.f;
  } else if (i < 16896) {
    int t = i - 12672; int n = t / 132, k = t % 132;
    dst[i] = (k < 130) ? Wsv[k * 32 + n] : 0.f;
  } else if (i < 23040) {
    int t = i - 16896; int n = t / 64, k = t % 64;
    dst[i] = Wvs[k * 96 + n];
  } else if (i < 25088) {
    int t = i - 23040; int n = t / 64, k = t % 64;
    dst[i] = Wvv0[k * 32 + n];
  } else if (i < 27136) {
    int t = i - 25088; int n = t / 64, k = t % 64;
    dst[i] = Wvv1[k * 32 + n];
  }
}

__global__ void zero_kernel(float* __restrict__ p, size_t n) {
  size_t i = (size_t)blockIdx.x * blockDim.x + threadIdx.x;
  if (i < n) p[i] = 0.f;
}

// ---------------------------------------------------------------------------
// Embed: s = (x@Wss)*na0 + bs ; v = (x@Wsv) outer na1
// ---------------------------------------------------------------------------
__global__ void embed_kernel(const float* __restrict__ x, const float* __restrict__ nattr,
                             const float* __restrict__ Wss, const float* __restrict__ Wsv,
                             const float* __restrict__ bs, float* __restrict__ s,
                             float* __restrict__ v, int Nn) {
  size_t idx = (size_t)blockIdx.x * blockDim.x + threadIdx.x;
  if (idx >= (size_t)Nn * 96) return;
  int n = (int)(idx / 96), o = (int)(idx % 96);
  const float* xr = x + (size_t)n * 15;
  if (o < 64) {
    float acc = 0.f;
    #pragma unroll
    for (int i = 0; i < 15; ++i) acc += xr[i] * Wss[i * 64 + o];
    s[(size_t)n * H0C + o] = acc * nattr[n * 4] + bs[o];
  } else {
    int j = o - 64;
    float acc = 0.f;
    #pragma unroll
    for (int i = 0; i < 15; ++i) acc += xr[i] * Wsv[i * 32 + j];
    size_t vb = ((size_t)n * H1C + j) * 3;
    v[vb + 0] = acc * nattr[n * 4 + 1];
    v[vb + 1] = acc * nattr[n * 4 + 2];
    v[vb + 2] = acc * nattr[n * 4 + 3];
  }
}

// ---------------------------------------------------------------------------
// Edge layer: one wave = 16 edges (M tile). fp32 WMMA, branch-free hot loops.
// ---------------------------------------------------------------------------
__global__ __launch_bounds__(128)
void edge_kernel(const float* __restrict__ sN, const float* __restrict__ vN,
                 const float* __restrict__ eattr, const float* __restrict__ amf,
                 const int* __restrict__ ei, int E,
                 const float* __restrict__ WT, const float* __restrict__ bs,
                 float* __restrict__ ags, float* __restrict__ agv) {
  const int wave = (int)(((size_t)blockIdx.x * blockDim.x + threadIdx.x) >> 5);
  const int lane = threadIdx.x & 31;
  const int ntile = (E + 15) >> 4;
  if (wave >= ntile) return;
  const int m = lane & 15;
  const int half = lane >> 4;
  const int kh = half * 2;

  const int e = wave * 16 + m;
  const bool valid = (e < E);
  const int ec = valid ? e : (E - 1);      // clamped, always in-bounds
  const float vmask = valid ? 1.f : 0.f;
  const int srcn = ei[ec];
  const int dstn = ei[E + ec];
  // unconditional loads + arithmetic masking (no EXEC divergence)
  const float ea0 = eattr[(size_t)ec * 4 + 0] * vmask;
  const float e1x = eattr[(size_t)ec * 4 + 1] * vmask;
  const float e1y = eattr[(size_t)ec * 4 + 2] * vmask;
  const float e1z = eattr[(size_t)ec * 4 + 3] * vmask;
  const float amf0 = amf[(size_t)ec * 2 + 0] * vmask;
  const float amf1 = amf[(size_t)ec * 2 + 1] * vmask;

  const float* __restrict__ sd  = sN + (size_t)dstn * H0C;
  const float* __restrict__ ssr = sN + (size_t)srcn * H0C;
  const float* __restrict__ vd  = vN + (size_t)dstn * (H1C * 3);
  const float* __restrict__ vs  = vN + (size_t)srcn * (H1C * 3);

  const float* __restrict__ WssT = WT;                 // [96][132]
  const float* __restrict__ WsvT = WT + 96 * 132;      // [32][132]
  const float* __restrict__ WvsT = WsvT + 32 * 132;    // [96][64]
  const float* __restrict__ W0T  = WvsT + 96 * 64;     // [32][64]
  const float* __restrict__ W1T  = W0T + 32 * 64;      // [32][64]

  const v8f z = {0.f, 0.f, 0.f, 0.f, 0.f, 0.f, 0.f, 0.f};
  v8f accS[6] = {z, z, z, z, z, z};
  v8f accT0[2] = {z, z};
  v8f aVx[2] = {z, z}, aVy[2] = {z, z}, aVz[2] = {z, z};

  // ---- scalar path, columns 0..63: s[dst] ----
  #pragma unroll 4
  for (int k = 0; k < 64; k += 4) {
    const int kl = k + kh;
    const v2f A = *(const v2f*)(sd + kl);
    step_scalar(m, kl, A, A * ea0, WssT, WsvT, accS, accT0);
  }
  // ---- scalar path, columns 64..127: s[src] ----
  #pragma unroll 4
  for (int k = 64; k < 128; k += 4) {
    const int kl = k + kh;
    const v2f A = *(const v2f*)(ssr + (kl - 64));
    step_scalar(m, kl, A, A * ea0, WssT, WsvT, accS, accT0);
  }
  // ---- scalar path tail, columns 128..131: amf | zero-pad ----
  {
    const float hsel = half ? 0.f : 1.f;   // half 1 covers padded cols 130,131
    const v2f A = {amf0 * hsel, amf1 * hsel};
    step_scalar(m, 128 + kh, A, A * ea0, WssT, WsvT, accS, accT0);
  }

  // ---- vector path, channels 0..31: v[dst] ----
  #pragma unroll 2
  for (int k = 0; k < 64; k += 4) {
    const int j0 = k + kh;
    const float* p = (k < 32) ? (vd + j0 * 3) : (vs + (j0 - 32) * 3);
    // NOTE: k is a compile-constant after unroll split below; keep branchless by
    // two explicit loops instead:
    (void)p;
    break;
  }
  #pragma unroll 2
  for (int k = 0; k < 32; k += 4) {
    const int j0 = k + kh;
    const float* p0 = vd + j0 * 3;
    const float vx0 = p0[0], vy0 = p0[1], vz0 = p0[2];
    const float vx1 = p0[3], vy1 = p0[4], vz1 = p0[5];
    const float d0 = vx0 * e1x + vy0 * e1y + vz0 * e1z;
    const float d1 = vx1 * e1x + vy1 * e1y + vz1 * e1z;
    const v2f Ad = {d0, d1};
    const v2f Ax = {vx0 * ea0, vx1 * ea0}, Ay = {vy0 * ea0, vy1 * ea0}, Az = {vz0 * ea0, vz1 * ea0};
    const v2f Cx = {vy0 * e1z - vz0 * e1y, vy1 * e1z - vz1 * e1y};
    const v2f Cy = {vz0 * e1x - vx0 * e1z, vz1 * e1x - vx1 * e1z};
    const v2f Cz = {vx0 * e1y - vy0 * e1x, vx1 * e1y - vy1 * e1x};
    step_vector(m, j0, Ad, Ax, Ay, Az, Cx, Cy, Cz, WvsT, W0T, W1T, accS, aVx, aVy, aVz);
  }
  // ---- vector path, channels 32..63: v[src] ----
  #pragma unroll 2
  for (int k = 32; k < 64; k += 4) {
    const int j0 = k + kh;
    const float* p0 = vs + (j0 - 32) * 3;
    const float vx0 = p0[0], vy0 = p0[1], vz0 = p0[2];
    const float vx1 = p0[3], vy1 = p0[4], vz1 = p0[5];
    const float d0 = vx0 * e1x + vy0 * e1y + vz0 * e1z;
    const float d1 = vx1 * e1x + vy1 * e1y + vz1 * e1z;
    const v2f Ad = {d0, d1};
    const v2f Ax = {vx0 * ea0, vx1 * ea0}, Ay = {vy0 * ea0, vy1 * ea0}, Az = {vz0 * ea0, vz1 * ea0};
    const v2f Cx = {vy0 * e1z - vz0 * e1y, vy1 * e1z - vz1 * e1y};
    const v2f Cy = {vz0 * e1x - vx0 * e1z, vz1 * e1x - vx1 * e1z};
    const v2f Cz = {vx0 * e1y - vy0 * e1x, vx1 * e1y - vy1 * e1x};
    step_vector(m, j0, Ad, Ax, Ay, Az, Cx, Cy, Cz, WvsT, W0T, W1T, accS, aVx, aVy, aVz);
  }

  // ---- epilogue: scatter-add into segment sums ----
  const float is2 = 0.70710678118654752f;
  const float is3 = 0.57735026918962576f;
  float bias[6];
  #pragma unroll
  for (int nt = 0; nt < 6; ++nt) bias[nt] = bs[nt * 16 + m];

  #pragma unroll
  for (int r = 0; r < 8; ++r) {
    const int row = r + half * 8;            // C-fragment row for VGPR r
    const int er = wave * 16 + row;
    const int dr = __shfl(dstn, row, 32);
    const float rx = __shfl(e1x, row, 32);
    const float ry = __shfl(e1y, row, 32);
    const float rz = __shfl(e1z, row, 32);
    if (er < E) {
      #pragma unroll
      for (int nt = 0; nt < 6; ++nt)
        atomicAdd(&ags[(size_t)dr * 96 + nt * 16 + m], accS[nt][r] * is2 + bias[nt]);
      #pragma unroll
      for (int nt = 0; nt < 2; ++nt) {
        const float t0 = accT0[nt][r];
        const size_t vb = ((size_t)dr * H1C + nt * 16 + m) * 3;
        atomicAdd(&agv[vb + 0], (t0 * rx + aVx[nt][r]) * is3);
        atomicAdd(&agv[vb + 1], (t0 * ry + aVy[nt][r]) * is3);
        atomicAdd(&agv[vb + 2], (t0 * rz + aVz[nt][r]) * is3);
      }
    }
  }
}

// ---------------------------------------------------------------------------
// Node update: s += silu(ags[:,:64]); v += agv * sigmoid(ags[:,64:]).
// Re-zeroes ags/agv for the next layer.
// ---------------------------------------------------------------------------
__global__ void update_kernel(float* __restrict__ s, float* __restrict__ v,
                              float* __restrict__ ags, float* __restrict__ agv, int Nn) {
  size_t idx = (size_t)blockIdx.x * blockDim.x + threadIdx.x;
  if (idx >= (size_t)Nn * 96) return;
  int n = (int)(idx / 96), o = (int)(idx % 96);
  float a = ags[idx];
  ags[idx] = 0.f;
  if (o < 64) {
    s[(size_t)n * H0C + o] += a / (1.f + __expf(-a));
  } else {
    int j = o - 64;
    float g = 1.f / (1.f + __expf(-a));
    size_t vb = ((size_t)n * H1C + j) * 3;
    #pragma unroll
    for (int c = 0; c < 3; ++c) {
      float t = agv[vb + c];
      agv[vb + c] = 0.f;
      v[vb + c] += t * g;
    }
  }
}

// ---------------------------------------------------------------------------
// Readout: out1 TP -> swish gate -> out2 TP. One wave per node.
// ---------------------------------------------------------------------------
__global__ __launch_bounds__(256)
void out_kernel(const float* __restrict__ sN, const float* __restrict__ vN,
                const float* __restrict__ nattr,
                const float* __restrict__ W1ss, const float* __restrict__ W1sv,
                const float* __restrict__ W1vs, const float* __restrict__ W1vv0,
                const float* __restrict__ W1vv1, const float* __restrict__ b1,
                const float* __restrict__ W2ss, const float* __restrict__ W2vs,
                const float* __restrict__ b2, float* __restrict__ out, int Nn) {
  const int wave = (int)(((size_t)blockIdx.x * blockDim.x + threadIdx.x) >> 5);
  const int lane = threadIdx.x & 31;
  if (wave >= Nn) return;
  const int n = wave;
  const float a0 = nattr[n * 4 + 0];
  const float a1x = nattr[n * 4 + 1], a1y = nattr[n * 4 + 2], a1z = nattr[n * 4 + 3];

  const size_t vb = ((size_t)n * H1C + lane) * 3;
  const float vx = vN[vb + 0], vy = vN[vb + 1], vz = vN[vb + 2];
  const float dj = vx * a1x + vy * a1y + vz * a1z;
  const float crx = vy * a1z - vz * a1y;
  const float cry = vz * a1x - vx * a1z;
  const float crz = vx * a1y - vy * a1x;

  const float* srow = sN + (size_t)n * H0C;
  float g0 = 0.f, g1 = 0.f, g2 = 0.f, t0 = 0.f;
  for (int k = 0; k < 64; ++k) {
    const float sk = srow[k];
    g0 += sk * W1ss[k * 96 + lane];
    g1 += sk * W1ss[k * 96 + lane + 32];
    g2 += sk * W1ss[k * 96 + lane + 64];
    t0 += sk * W1sv[k * 32 + lane];
  }
  float h0 = 0.f, h1 = 0.f, h2 = 0.f;
  float u1x = 0.f, u1y = 0.f, u1z = 0.f, u2x = 0.f, u2y = 0.f, u2z = 0.f;
  for (int i = 0; i < 32; ++i) {
    const float djj = __shfl(dj, i, 32);
    h0 += djj * W1vs[i * 96 + lane];
    h1 += djj * W1vs[i * 96 + lane + 32];
    h2 += djj * W1vs[i * 96 + lane + 64];
    const float vxi = __shfl(vx, i, 32), vyi = __shfl(vy, i, 32), vzi = __shfl(vz, i, 32);
    const float cxi = __shfl(crx, i, 32), cyi = __shfl(cry, i, 32), czi = __shfl(crz, i, 32);
    const float w0 = W1vv0[i * 32 + lane], w1 = W1vv1[i * 32 + lane];
    u1x += vxi * w0; u1y += vyi * w0; u1z += vzi * w0;
    u2x += cxi * w1; u2y += cyi * w1; u2z += czi * w1;
  }
  const float is2 = 0.70710678118654752f;
  const float is3 = 0.57735026918962576f;
  const float gs0 = (g0 * a0 + h0) * is2 + b1[lane];
  const float gs1 = (g1 * a0 + h1) * is2 + b1[lane + 32];
  const float gs2 = (g2 * a0 + h2) * is2 + b1[lane + 64];
  const float gvx = (t0 * a1x + u1x * a0 + u2x) * is3;
  const float gvy = (t0 * a1y + u1y * a0 + u2y) * is3;
  const float gvz = (t0 * a1z + u1z * a0 + u2z) * is3;

  const float sp0 = gs0 / (1.f + __expf(-gs0));
  const float sp1 = gs1 / (1.f + __expf(-gs1));
  const float gate = 1.f / (1.f + __expf(-gs2));
  const float wx = gvx * gate, wy = gvy * gate, wz = gvz * gate;

  float ps = sp0 * W2ss[lane] + sp1 * W2ss[lane + 32];
  float pv = (wx * a1x + wy * a1y + wz * a1z) * W2vs[lane];
  #pragma unroll
  for (int off = 16; off > 0; off >>= 1) {
    ps += __shfl_xor(ps, off, 32);
    pv += __shfl_xor(pv, off, 32);
  }
  if (lane == 0) out[n] = (ps * a0 + pv) * is2 + b2[0];
}

// ---------------------------------------------------------------------------
extern "C" void kernel_launch(void* const* d_in, const int* in_sizes, int n_in,
                              void* d_out, int out_size, void* d_ws, size_t ws_size,
                              hipStream_t stream) {
  const float* x     = (const float*)d_in[0];
  const float* nattr = (const float*)d_in[1];
  const float* eattr = (const float*)d_in[2];
  const float* amf   = (const float*)d_in[3];
  const int*   ei    = (const int*)d_in[4];
  // d_in[5] = batch (unused)
  const int Nn = in_sizes[0] / 15;
  const int E  = in_sizes[4] / 2;

  auto F = [&](int i) { return (const float*)d_in[i]; };
  // params flattened (jax dict order, keys sorted):
  // 6..8   embed:  Wss[15,64], Wsv[15,32], bs[64]
  // 9+6l.. layer l: Wss[130,96], Wsv[130,32], Wvs[64,96], Wvv0[64,32], Wvv1[64,32], bs[96]
  // 33..38 out1:   Wss[64,96], Wsv[64,32], Wvs[32,96], Wvv0[32,32], Wvv1[32,32], bs[96]
  // 39..41 out2:   Wss[64,1], Wvs[32,1], bs[1]

  float* ws = (float*)d_ws;
  size_t o = 0;
  float* sBuf = ws + o; o += (size_t)Nn * 64;
  float* vBuf = ws + o; o += (size_t)Nn * 96;
  float* ags  = ws + o; o += (size_t)Nn * 96;
  float* agv  = ws + o; o += (size_t)Nn * 96;
  float* wt   = ws + o;                      // 4 * 27136 floats
  const int WT_SZ = 27136;

  for (int l = 0; l < 4; ++l) {
    int b = 9 + l * 6;
    prep_weights<<<(WT_SZ + 255) / 256, 256, 0, stream>>>(
        F(b), F(b + 1), F(b + 2), F(b + 3), F(b + 4), wt + (size_t)l * WT_SZ);
  }
  const size_t nz = (size_t)Nn * 192;  // ags + agv are contiguous
  zero_kernel<<<(int)((nz + 255) / 256), 256, 0, stream>>>(ags, nz);

  const size_t ne = (size_t)Nn * 96;
  embed_kernel<<<(int)((ne + 255) / 256), 256, 0, stream>>>(
      x, nattr, F(6), F(7), F(8), sBuf, vBuf, Nn);

  const int tiles = (E + 15) / 16;
  const int eblocks = (tiles * 32 + 127) / 128;
  for (int l = 0; l < 4; ++l) {
    int b = 9 + l * 6;
    edge_kernel<<<eblocks, 128, 0, stream>>>(
        sBuf, vBuf, eattr, amf, ei, E, wt + (size_t)l * WT_SZ, F(b + 5), ags, agv);
    update_kernel<<<(int)((ne + 255) / 256), 256, 0, stream>>>(sBuf, vBuf, ags, agv, Nn);
  }

  out_kernel<<<(Nn * 32 + 255) / 256, 256, 0, stream>>>(
      sBuf, vBuf, nattr, F(33), F(34), F(35), F(36), F(37), F(38),
      F(39), F(40), F(41), (float*)d_out, Nn);
}